// PixelVectorExtractor_56427280335181
// MI455X (gfx1250) — compile-verified
//
#include <hip/hip_runtime.h>

// Problem constants (from reference)
#define NB   2
#define CC   10
#define HH   30
#define WW   30
#define DD   11          // D = C+1
#define LL   100         // L = 10*10
#define LP   112         // L padded to 7 * 16 WMMA M-tiles
#define SD   12          // channel stride: D padded to 3 K-steps of 4
#define BTOT (NB*HH*WW)  // 1800 pixels
#define WAVES 2          // waves (pixels) per block

typedef __attribute__((ext_vector_type(2))) float v2f;
typedef __attribute__((ext_vector_type(8))) float v8f;

__device__ __forceinline__ v8f wmma_f32_k4(v2f a, v2f b, v8f c) {
  // V_WMMA_F32_16X16X4_F32 : D = A(16x4) * B(4x16) + C(16x16), fp32
  return __builtin_amdgcn_wmma_f32_16x16x4_f32(
      /*neg_a=*/false, a, /*neg_b=*/false, b,
      /*c_mod=*/(short)0, c, /*reuse_a=*/false, /*reuse_b=*/false);
}

__global__ void __launch_bounds__(WAVES * 32)
pve_attention_kernel(const float* __restrict__ x,
                     const float* __restrict__ w_in,
                     const float* __restrict__ w_out,
                     const float* __restrict__ w_ff1,
                     const float* __restrict__ w_ff2,
                     const float* __restrict__ ln1_g,
                     const float* __restrict__ ln2_g,
                     float* __restrict__ out)
{
  // Shared weights (block-wide)
  __shared__ float s_winT[12 * 48];   // w_in^T  : [K=12 pad][N=48 pad of 33]
  __shared__ float s_woutT[12 * 16];  // w_out^T : [K=12 pad][N=16 pad of 11]
  __shared__ float s_ff1[16], s_ff2[16], s_g1[16], s_g2[16];
  // Per-wave working set
  __shared__ float s_seq[WAVES][LP * SD];
  __shared__ float s_q  [WAVES][LP * SD];   // later reused for proj = ao @ w_out^T
  __shared__ float s_k  [WAVES][LP * SD];
  __shared__ float s_v  [WAVES][LP * SD];
  __shared__ float s_ao [WAVES][LP * SD];

  const int tid  = threadIdx.x;
  const int lane = tid & 31;
  const int wv   = tid >> 5;
  const int b    = blockIdx.x * WAVES + wv;   // pixel id, grid sized exactly

  // ---- cooperative weight staging ----
  for (int idx = tid; idx < 12 * 48; idx += WAVES * 32) {
    const int k = idx / 48, nn = idx % 48;
    s_winT[idx] = (k < DD && nn < 3 * DD) ? w_in[nn * DD + k] : 0.f;
  }
  for (int idx = tid; idx < 12 * 16; idx += WAVES * 32) {
    const int k = idx / 16, nn = idx % 16;
    s_woutT[idx] = (k < DD && nn < DD) ? w_out[nn * DD + k] : 0.f;
  }
  if (tid < DD) {
    s_ff1[tid] = w_ff1[tid];
    s_ff2[tid] = w_ff2[tid];
    s_g1[tid]  = ln1_g[tid];
    s_g2[tid]  = ln2_g[tid];
  }

  float* seq = s_seq[wv];
  float* qA  = s_q[wv];
  float* kA  = s_k[wv];
  float* vA  = s_v[wv];
  float* ao  = s_ao[wv];

  // ---- phase 0: zero pads + build the 100x11 patch sequence ----
  for (int t = lane; t < LP * SD; t += 32) { seq[t] = 0.f; ao[t] = 0.f; }

  const int n  = b / (HH * WW);
  const int ij = b % (HH * WW);
  const int ci = ij / WW;
  const int cj = ij % WW;

  for (int l = lane; l < LL; l += 32) {
    const int ph = l / 10, pw = l % 10;
    if (ph < 9 && pw < 9) {                     // inside 9x9 window
      const int gi = ci + ph, gj = cj + pw;     // coords in padded 38x38
      if (gi >= 4 && gi < 34 && gj >= 4 && gj < 34) {
        const int yy = gi - 4, xx = gj - 4;
        #pragma unroll
        for (int c = 0; c < CC; ++c)
          seq[l * SD + c] = x[((n * CC + c) * HH + yy) * WW + xx];
      } else {
        seq[l * SD + CC] = 1.f;                 // border indicator channel
      }
    }
  }
  __syncthreads();

  const int mrow  = lane & 15;
  const int khalf = (lane >> 4) << 1;           // 0 (lanes 0-15) or 2 (16-31)
  const int mhalf = (lane >> 4) << 3;           // C-tile row offset 0 or 8

  // ---- phase 1: qkv = seq @ w_in^T  (M=112, N=48, K=12) via f32 WMMA ----
  for (int mt = 0; mt < 7; ++mt) {
    for (int nt = 0; nt < 3; ++nt) {
      v8f acc = {};
      #pragma unroll
      for (int kt = 0; kt < 3; ++kt) {
        const int kb = kt * 4 + khalf;
        v2f a, bb;
        a.x  = seq[(mt * 16 + mrow) * SD + kb];
        a.y  = seq[(mt * 16 + mrow) * SD + kb + 1];
        bb.x = s_winT[kb * 48 + nt * 16 + mrow];
        bb.y = s_winT[(kb + 1) * 48 + nt * 16 + mrow];
        acc  = wmma_f32_k4(a, bb, acc);
      }
      // scatter C tile into q / k / v (predicated per lane)
      const int nn    = nt * 16 + mrow;
      const int mbase = mt * 16 + mhalf;
      float* dst = nullptr; int col = 0;
      if      (nn <     DD) { dst = qA; col = nn;          }
      else if (nn < 2 * DD) { dst = kA; col = nn - DD;     }
      else if (nn < 3 * DD) { dst = vA; col = nn - 2 * DD; }
      if (dst) {
        #pragma unroll
        for (int r = 0; r < 8; ++r)
          dst[(mbase + r) * SD + col] = acc[r];
      }
    }
  }
  __syncthreads();

  // ---- phase 2: per-channel attention (softmax over the 81 valid k) ----
  for (int d = 0; d < DD; ++d) {
    for (int qc = 0; qc < 4; ++qc) {
      const int qi = qc * 32 + lane;
      const float qv = (qi < LL) ? qA[qi * SD + d] : 0.f;
      float mx = -1e30f;
      for (int kh = 0; kh < 9; ++kh)
        for (int kw = 0; kw < 9; ++kw)
          mx = fmaxf(mx, qv * kA[(kh * 10 + kw) * SD + d]);
      float ssum = 0.f, accv = 0.f;
      for (int kh = 0; kh < 9; ++kh)
        for (int kw = 0; kw < 9; ++kw) {
          const int kl = kh * 10 + kw;
          const float e = __expf(qv * kA[kl * SD + d] - mx);
          ssum += e;
          accv += e * vA[kl * SD + d];
        }
      if (qi < LL) ao[qi * SD + d] = accv / ssum;
    }
  }
  __syncthreads();

  // ---- phase 3: proj = ao @ w_out^T  (M=112, N=16, K=12), writes into qA ----
  for (int mt = 0; mt < 7; ++mt) {
    v8f acc = {};
    #pragma unroll
    for (int kt = 0; kt < 3; ++kt) {
      const int kb = kt * 4 + khalf;
      v2f a, bb;
      a.x  = ao[(mt * 16 + mrow) * SD + kb];
      a.y  = ao[(mt * 16 + mrow) * SD + kb + 1];
      bb.x = s_woutT[kb * 16 + mrow];
      bb.y = s_woutT[(kb + 1) * 16 + mrow];
      acc  = wmma_f32_k4(a, bb, acc);
    }
    const int mbase = mt * 16 + mhalf;
    if (mrow < DD) {
      #pragma unroll
      for (int r = 0; r < 8; ++r)
        qA[(mbase + r) * SD + mrow] = acc[r];
    }
  }
  __syncthreads();

  // ---- phase 4: LN1 -> FFN -> LN2 -> channel softmax -> output (row-local) ----
  for (int qc = 0; qc < 4; ++qc) {
    const int r = qc * 32 + lane;
    if (r >= LL) continue;
    float sv[DD], tv[DD];
    float mean = 0.f;
    #pragma unroll
    for (int d = 0; d < DD; ++d) {
      sv[d] = seq[r * SD + d];
      tv[d] = sv[d] + qA[r * SD + d];
      mean += tv[d];
    }
    mean *= (1.f / 11.f);
    float var = 0.f;
    #pragma unroll
    for (int d = 0; d < DD; ++d) { const float df = tv[d] - mean; var += df * df; }
    var *= (1.f / 11.f);
    const float inv1 = rsqrtf(var + 1e-5f);
    float h1[DD];
    float dot = 0.f;
    #pragma unroll
    for (int d = 0; d < DD; ++d) {
      h1[d] = (tv[d] - mean) * inv1 * s_g1[d];
      dot  += h1[d] * s_ff1[d];
    }
    const float rl = fmaxf(dot, 0.f);
    float u[DD]; float mean2 = 0.f;
    #pragma unroll
    for (int d = 0; d < DD; ++d) { u[d] = h1[d] + rl * s_ff2[d]; mean2 += u[d]; }
    mean2 *= (1.f / 11.f);
    float var2 = 0.f;
    #pragma unroll
    for (int d = 0; d < DD; ++d) { const float df = u[d] - mean2; var2 += df * df; }
    var2 *= (1.f / 11.f);
    const float inv2 = rsqrtf(var2 + 1e-5f);
    float h2[CC];
    float mx = -1e30f;
    #pragma unroll
    for (int c = 0; c < CC; ++c) {
      h2[c] = (u[c] - mean2) * inv2 * s_g2[c];
      mx = fmaxf(mx, h2[c]);
    }
    float es = 0.f; float ev[CC];
    #pragma unroll
    for (int c = 0; c < CC; ++c) { ev[c] = __expf(h2[c] - mx); es += ev[c]; }
    const float mask0 = (sv[CC] == 1.f) ? 1.f : 0.f;
    const float scale = mask0 / es;
    #pragma unroll
    for (int c = 0; c < CC; ++c)
      out[(b * CC + c) * LL + r] = sv[c] + ev[c] * scale;
  }
}

extern "C" void kernel_launch(void* const* d_in, const int* in_sizes, int n_in,
                              void* d_out, int out_size, void* d_ws, size_t ws_size,
                              hipStream_t stream) {
  (void)in_sizes; (void)n_in; (void)out_size; (void)d_ws; (void)ws_size;
  const float* x     = (const float*)d_in[0];
  const float* w_in  = (const float*)d_in[1];
  const float* w_out = (const float*)d_in[2];
  const float* w_ff1 = (const float*)d_in[3];
  const float* w_ff2 = (const float*)d_in[4];
  const float* g1    = (const float*)d_in[5];
  const float* g2    = (const float*)d_in[6];
  dim3 grid(BTOT / WAVES);   // 900 blocks, 2 waves (pixels) each
  dim3 block(WAVES * 32);
  pve_attention_kernel<<<grid, block, 0, stream>>>(
      x, w_in, w_out, w_ff1, w_ff2, g1, g2, (float*)d_out);
}